// GraphAttentionModel_41291815584007
// MI455X (gfx1250) — compile-verified
//
#include <hip/hip_runtime.h>

typedef float v2f __attribute__((ext_vector_type(2)));
typedef float v8f __attribute__((ext_vector_type(8)));

#define NEG_SLOPE 0.2f

// ---------------------------------------------------------------------------
// Monotone uint key for float so atomicMax(uint) == float max.
// ---------------------------------------------------------------------------
__device__ __forceinline__ unsigned fkey(float f) {
    unsigned b = __float_as_uint(f);
    return b ^ ((unsigned)((int)b >> 31) | 0x80000000u);
}
__device__ __forceinline__ float fdekey(unsigned k) {
    unsigned b = (k & 0x80000000u) ? (k ^ 0x80000000u) : ~k;
    return __uint_as_float(b);
}

// ---------------------------------------------------------------------------
// Kernel 1: H = X @ W   (X: n x 64 row-major, W: 64 x 64 row-major)
// fp32 WMMA 16x16x4. Block = 128 threads = 4 waves; wave w owns the 16-wide
// column tile [16w, 16w+16); block owns 16 rows. K loop: 16 steps of 4.
// A-frag (16x4 f32): lane L holds A[L%16][2*(L/16)+v], v = vgpr 0/1.
// B-frag (4x16 f32): lane L holds B[2*(L/16)+v][L%16].
// C/D (16x16 f32):  lane L, vgpr r holds C[r + 8*(L/16)][L%16].
// ---------------------------------------------------------------------------
__global__ __launch_bounds__(128) void gat_gemm64(
    const float* __restrict__ X, const float* __restrict__ W,
    float* __restrict__ H, int n)
{
    const int m0   = blockIdx.x << 4;
    const int wv   = threadIdx.x >> 5;     // 0..3 : column tile
    const int lane = threadIdx.x & 31;
    const int lm   = lane & 15;
    const int lh   = lane >> 4;            // K-half selector
    const int n0   = wv << 4;

    int row = m0 + lm;
    if (row >= n) row = n - 1;             // clamp (n is a multiple of 16 here)
    const float* xr = X + (size_t)row * 64;

    v8f c = {};
#pragma unroll
    for (int k = 0; k < 64; k += 4) {
        v2f a, b;
        a.x = xr[k + 2 * lh + 0];
        a.y = xr[k + 2 * lh + 1];
        b.x = W[(k + 2 * lh + 0) * 64 + n0 + lm];
        b.y = W[(k + 2 * lh + 1) * 64 + n0 + lm];
        c = __builtin_amdgcn_wmma_f32_16x16x4_f32(
                false, a, false, b, (short)0, c, false, false);
    }
#pragma unroll
    for (int r = 0; r < 8; ++r) {
        int orow = m0 + r + 8 * lh;
        if (orow < n) H[(size_t)orow * 64 + n0 + lm] = c[r];
    }
}

// ---------------------------------------------------------------------------
// Kernel 2: per-node attention scores  s = H.a_src, t = H.a_dst
// ---------------------------------------------------------------------------
__global__ void gat_st(const float* __restrict__ H,
                       const float* __restrict__ asrc,
                       const float* __restrict__ adst,
                       float* __restrict__ s, float* __restrict__ t, int n)
{
    int i = blockIdx.x * blockDim.x + threadIdx.x;
    if (i >= n) return;
    const float* hr = H + (size_t)i * 64;
    float ss = 0.f, tt = 0.f;
#pragma unroll 8
    for (int f = 0; f < 64; ++f) {
        float hv = hr[f];
        ss += hv * asrc[f];
        tt += hv * adst[f];
    }
    s[i] = ss;
    t[i] = tt;
}

// ---------------------------------------------------------------------------
// Kernel 3: zero accumulators
// ---------------------------------------------------------------------------
__global__ void gat_init(unsigned* __restrict__ mk, float* __restrict__ den,
                         float* __restrict__ acc, int n)
{
    int i = blockIdx.x * blockDim.x + threadIdx.x;
    if (i < n * 64) acc[i] = 0.f;
    if (i < n) { mk[i] = 0u; den[i] = 0.f; }
}

// ---------------------------------------------------------------------------
// Kernel 4: segment max over incoming edges (incl. self loops) via uint key
// ---------------------------------------------------------------------------
__global__ void gat_max(const long long* __restrict__ ei,
                        const float* __restrict__ s, const float* __restrict__ t,
                        unsigned* __restrict__ mk, int E, int n)
{
    int i = blockIdx.x * blockDim.x + threadIdx.x;
    if (i >= E + n) return;
    long long src, dst;
    if (i < E) { src = ei[i]; dst = ei[E + i]; }
    else       { src = dst = (long long)(i - E); }
    float e = s[src] + t[dst];
    e = (e >= 0.f) ? e : NEG_SLOPE * e;
    atomicMax(mk + dst, fkey(e));
}

// ---------------------------------------------------------------------------
// Kernel 5: one wave per edge: ex = exp(e - m[dst]); accumulate denominator
// and ex * h[src] into acc[dst] with hardware fp32 atomics.
// ---------------------------------------------------------------------------
__global__ __launch_bounds__(256) void gat_agg(
    const long long* __restrict__ ei,
    const float* __restrict__ s, const float* __restrict__ t,
    const unsigned* __restrict__ mk, const float* __restrict__ H,
    float* __restrict__ den, float* __restrict__ acc, int E, int n)
{
    int eid  = blockIdx.x * 8 + (threadIdx.x >> 5);   // wave id == edge id
    int lane = threadIdx.x & 31;
    if (eid >= E + n) return;
    long long src, dst;
    if (eid < E) { src = ei[eid]; dst = ei[E + eid]; }
    else         { src = dst = (long long)(eid - E); }

    float e = s[src] + t[dst];
    e = (e >= 0.f) ? e : NEG_SLOPE * e;
    float m  = fdekey(mk[dst]);
    float ex = __expf(e - m);

    if (lane == 0) unsafeAtomicAdd(den + dst, ex);

    const float* hs = H + (size_t)src * 64;
    float* ad = acc + (size_t)dst * 64;
    unsafeAtomicAdd(ad + lane,      hs[lane]      * ex);
    unsafeAtomicAdd(ad + lane + 32, hs[lane + 32] * ex);
}

// ---------------------------------------------------------------------------
// Kernel 6: out = relu(acc / den + bias)
// ---------------------------------------------------------------------------
__global__ void gat_fin(const float* __restrict__ acc,
                        const float* __restrict__ den,
                        const float* __restrict__ bias,
                        float* __restrict__ out, int n)
{
    int i = blockIdx.x * blockDim.x + threadIdx.x;
    if (i >= n * 64) return;
    int node = i >> 6, f = i & 63;
    float v = acc[i] / den[node] + bias[f];
    out[i] = v > 0.f ? v : 0.f;
}

// ---------------------------------------------------------------------------
// Host launcher
// d_in: 0=x, 1=edge_index(int64, [2,E]),
//       then per layer l in {1,2,3}: W, a_src, a_dst, b at 2+4*(l-1)..
// ---------------------------------------------------------------------------
extern "C" void kernel_launch(void* const* d_in, const int* in_sizes, int n_in,
                              void* d_out, int out_size, void* d_ws, size_t ws_size,
                              hipStream_t stream)
{
    const float*     x  = (const float*)d_in[0];
    const long long* ei = (const long long*)d_in[1];
    const int N = in_sizes[0] / 64;
    const int E = in_sizes[1] / 2;

    // Workspace layout (floats): h, acc, bufA, bufB (N*64 each), s, t, mk, den (N each)
    float*    h    = (float*)d_ws;
    float*    acc  = h    + (size_t)N * 64;
    float*    bufA = acc  + (size_t)N * 64;
    float*    bufB = bufA + (size_t)N * 64;
    float*    s    = bufB + (size_t)N * 64;
    float*    t    = s + N;
    unsigned* mk   = (unsigned*)(t + N);
    float*    den  = (float*)(mk + N);

    const int THR = 256;
    const int nfBlocks   = (N * 64 + THR - 1) / THR;
    const int nBlocks    = (N + THR - 1) / THR;
    const int edgeBlocks = (E + N + THR - 1) / THR;
    const int aggBlocks  = (E + N + 7) / 8;     // 8 waves (edges) per 256-thr block
    const int gemmBlocks = (N + 15) / 16;

    const float* lin = x;
    for (int layer = 0; layer < 3; ++layer) {
        const float* W  = (const float*)d_in[2 + 4 * layer];
        const float* as = (const float*)d_in[3 + 4 * layer];
        const float* ad = (const float*)d_in[4 + 4 * layer];
        const float* b  = (const float*)d_in[5 + 4 * layer];
        float* lout = (layer == 0) ? bufA : (layer == 1) ? bufB : (float*)d_out;

        gat_gemm64<<<gemmBlocks, 128, 0, stream>>>(lin, W, h, N);
        gat_st   <<<nBlocks,     THR, 0, stream>>>(h, as, ad, s, t, N);
        gat_init <<<nfBlocks,    THR, 0, stream>>>(mk, den, acc, N);
        gat_max  <<<edgeBlocks,  THR, 0, stream>>>(ei, s, t, mk, E, N);
        gat_agg  <<<aggBlocks,   THR, 0, stream>>>(ei, s, t, mk, h, den, acc, E, N);
        gat_fin  <<<nfBlocks,    THR, 0, stream>>>(acc, den, b, lout, N);

        lin = lout;
    }
}